// MoeAdapter_82197084111053
// MI455X (gfx1250) — compile-verified
//
#include <hip/hip_runtime.h>
#include <hip/hip_bf16.h>
#include <stddef.h>

#define EMBED 256
#define DOWN  64
#define NE    10
#define T_TOT (16 * 4096)        // 65536 tokens
#define NPAIR (T_TOT * 2)        // 131072 (token, k) pairs
#define TILE  16
#define PPB   128                // pairs per block: 8 waves x 16 rows
#define BLOCKS_UB (NPAIR / PPB + NE)   // 1034 upper bound on FFN blocks
#define GATE_SCALE 0.5f
#define CV_EPS 1e-10f

// padded LDS strides (bank = addr/4 % 64; chosen so 32-lane b64 reads are conflict-free)
#define W1T_STRIDE 260           // W1^T[h][d], h in [0,64), d in [0,256)
#define W2T_STRIDE 68            // W2^T[o][h], o in [0,256), h in [0,64)
#define HB_STRIDE  68            // h tile row stride

typedef float v2f __attribute__((ext_vector_type(2)));
typedef float v8f __attribute__((ext_vector_type(8)));

typedef __attribute__((address_space(3))) float       lds_float;
typedef __attribute__((address_space(1))) const float glb_cfloat;
typedef __attribute__((address_space(3))) int         lds_int;
typedef __attribute__((address_space(1))) int         glb_int;

__device__ __forceinline__ void stage_b32(const float* g, float* l) {
#if defined(__HIP_DEVICE_COMPILE__) && __has_builtin(__builtin_amdgcn_global_load_async_to_lds_b32)
    // async memory->LDS copy (ASYNCcnt), per-lane LDS destination does the transpose
    __builtin_amdgcn_global_load_async_to_lds_b32(
        (glb_int*)(glb_cfloat*)g,   // generic->AS1 addrspacecast, then ->int* (drop const)
        (lds_int*)(lds_float*)l,    // generic->AS3 addrspacecast, then ->int*
        0, 0);
#else
    *l = *g;
#endif
}

__device__ __forceinline__ void stage_wait() {
#if defined(__HIP_DEVICE_COMPILE__) && __has_builtin(__builtin_amdgcn_global_load_async_to_lds_b32)
#if __has_builtin(__builtin_amdgcn_s_wait_asynccnt)
    __builtin_amdgcn_s_wait_asynccnt(0);
#else
    asm volatile("s_wait_asynccnt 0x0" ::: "memory");
#endif
#endif
}

// ---------------- workspace layout (32-bit units) ----------------
// wsf[0..9]    importance (float atomics)     wsi[16..25] loadCnt (int atomics)
// wsi[32..42]  offsets[11]                    wsi[48..58] blkOff[11]
// wsi[64..73]  cursors[10]                    wsf[80] auxLoss   wsi[81] nBlocks
// wsi[128..)            tmpIdx [NPAIR]
// wsf[128+NPAIR..)      tmpGate[NPAIR]
// wsi[128+2*NPAIR..)    pairTok[NPAIR]
// wsf[128+3*NPAIR..)    pairGate[NPAIR]
// wsf[128+4*NPAIR..)    partials[4096]

// ---------------- zero init ----------------
__global__ __launch_bounds__(256) void moe_zero(float* __restrict__ out,
                                                float* __restrict__ importance,
                                                int* __restrict__ loadCnt) {
    size_t i = (size_t)blockIdx.x * 256 + threadIdx.x;
    const size_t n = (size_t)T_TOT * EMBED;
    if (i < n) out[i] = 0.0f;
    if (i < NE) { importance[i] = 0.0f; loadCnt[i] = 0; }
}

// ---------------- gating: one wave32 per token ----------------
__global__ __launch_bounds__(256) void moe_gate(const float* __restrict__ x,
                                                const float* __restrict__ wgate,
                                                int* __restrict__ tmpIdx,
                                                float* __restrict__ tmpGate,
                                                float* __restrict__ importance,
                                                int* __restrict__ loadCnt) {
    __shared__ float swg[EMBED * NE];
    for (int i = threadIdx.x; i < EMBED * NE; i += 256) swg[i] = wgate[i];
    __syncthreads();

    const int wave = threadIdx.x >> 5;
    const int lane = threadIdx.x & 31;
    const int t = blockIdx.x * 8 + wave;
    const float* xr = x + (size_t)t * EMBED;

    float lg[NE];
#pragma unroll
    for (int e = 0; e < NE; ++e) lg[e] = 0.0f;
#pragma unroll
    for (int i = 0; i < 8; ++i) {
        const int d = lane * 8 + i;
        const float xv = xr[d];
#pragma unroll
        for (int e = 0; e < NE; ++e) lg[e] += xv * swg[d * NE + e];
    }
#pragma unroll
    for (int e = 0; e < NE; ++e) {
#pragma unroll
        for (int off = 16; off > 0; off >>= 1)
            lg[e] += __shfl_xor(lg[e], off, 32);
    }
    // top-2 (ties -> lower index, matching lax.top_k)
    int e0 = 0; float v0 = lg[0];
#pragma unroll
    for (int e = 1; e < NE; ++e) if (lg[e] > v0) { v0 = lg[e]; e0 = e; }
    int e1 = (e0 == 0) ? 1 : 0; float v1 = lg[e1];
#pragma unroll
    for (int e = 0; e < NE; ++e)
        if (e != e0 && e != e1 && lg[e] > v1) { v1 = lg[e]; e1 = e; }
    const float ex = expf(v1 - v0);         // v1 <= v0, stable softmax
    const float g0 = 1.0f / (1.0f + ex);
    const float g1 = 1.0f - g0;

    if (lane == 0) {
        tmpIdx[t * 2 + 0] = e0;  tmpGate[t * 2 + 0] = g0;
        tmpIdx[t * 2 + 1] = e1;  tmpGate[t * 2 + 1] = g1;
        atomicAdd(importance + e0, g0);
        atomicAdd(importance + e1, g1);
        atomicAdd(loadCnt + e0, 1);
        atomicAdd(loadCnt + e1, 1);
    }
}

// ---------------- tiny scan + aux loss (1 thread) ----------------
__global__ void moe_scan(const float* __restrict__ importance,
                         const int* __restrict__ loadCnt,
                         int* __restrict__ offsets, int* __restrict__ blkOff,
                         int* __restrict__ cursors,
                         float* __restrict__ auxLoss, int* __restrict__ nBlocks) {
    if (threadIdx.x != 0 || blockIdx.x != 0) return;
    int off = 0, boff = 0;
    for (int e = 0; e < NE; ++e) {
        offsets[e] = off; blkOff[e] = boff; cursors[e] = 0;
        off += loadCnt[e];
        boff += (loadCnt[e] + PPB - 1) / PPB;
    }
    offsets[NE] = off; blkOff[NE] = boff; *nBlocks = boff;

    float mi = 0.0f, ml = 0.0f;
    for (int e = 0; e < NE; ++e) { mi += importance[e]; ml += (float)loadCnt[e]; }
    mi /= NE; ml /= NE;
    float vi = 0.0f, vl = 0.0f;
    for (int e = 0; e < NE; ++e) {
        const float di = importance[e] - mi; vi += di * di;
        const float dl = (float)loadCnt[e] - ml; vl += dl * dl;
    }
    vi /= (NE - 1); vl /= (NE - 1);   // unbiased variance (ddof=1)
    *auxLoss = vi / (mi * mi + CV_EPS) + vl / (ml * ml + CV_EPS);
}

// ---------------- scatter pairs into per-expert segments ----------------
__global__ __launch_bounds__(256) void moe_scatter(const int* __restrict__ tmpIdx,
                                                   const float* __restrict__ tmpGate,
                                                   const int* __restrict__ offsets,
                                                   int* __restrict__ cursors,
                                                   int* __restrict__ pairTok,
                                                   float* __restrict__ pairGate) {
    const int i = blockIdx.x * 256 + threadIdx.x;
    const int e = tmpIdx[i];
    const int pos = atomicAdd(cursors + e, 1);
    const int dst = offsets[e] + pos;
    pairTok[dst]  = i >> 1;
    pairGate[dst] = tmpGate[i];
}

// ---------------- expert FFN: 8 waves/block, 16-token tile per wave ----------------
// Per block: stage one expert's W1^T / W2^T into LDS (async mem->LDS), then every
// B operand of every WMMA is a single conflict-free ds_load_b64.
__global__ __launch_bounds__(256) void moe_ffn(const float* __restrict__ x,
                                               const float* __restrict__ w1,
                                               const float* __restrict__ b1,
                                               const float* __restrict__ w2,
                                               const float* __restrict__ b2,
                                               const int* __restrict__ pairTok,
                                               const float* __restrict__ pairGate,
                                               const int* __restrict__ offsets,
                                               const int* __restrict__ blkOff,
                                               const int* __restrict__ loadCnt,
                                               const int* __restrict__ nBlocks,
                                               float* __restrict__ out) {
    __shared__ float sW1T[DOWN * W1T_STRIDE];    // 66,560 B : W1^T[h][d]
    __shared__ float sW2T[EMBED * W2T_STRIDE];   // 69,632 B : W2^T[o][h]
    __shared__ float hbuf[8][TILE * HB_STRIDE];  // 34,816 B
    __shared__ int   stok[8][TILE];
    __shared__ float sg[8][TILE];

    const int blk = blockIdx.x;
    if (blk >= *nBlocks) return;        // block-uniform: EXEC stays full for WMMA

    int e = 0;
    while (blk >= blkOff[e + 1]) ++e;
    const int lb      = blk - blkOff[e];
    const int segBase = offsets[e];
    const int segCnt  = loadCnt[e];

    const float* w1e = w1 + (size_t)e * EMBED * DOWN;
    const float* w2e = w2 + (size_t)e * DOWN * EMBED;
    const float* b1e = b1 + e * DOWN;
    const float* b2e = b2 + e * EMBED;

    // ---- stage transposed weights into LDS (async; per-lane LDS dest = transpose) ----
    for (int it = threadIdx.x; it < EMBED * DOWN; it += 256) {
        const int d = it >> 6, h = it & 63;            // w1e[d][h] -> W1^T[h][d]
        stage_b32(w1e + it, &sW1T[h * W1T_STRIDE + d]);
    }
    for (int it = threadIdx.x; it < DOWN * EMBED; it += 256) {
        const int h = it >> 8, o = it & 255;           // w2e[h][o] -> W2^T[o][h]
        stage_b32(w2e + it, &sW2T[o * W2T_STRIDE + h]);
    }

    const int wave = threadIdx.x >> 5;
    const int lane = threadIdx.x & 31;
    const int half = lane >> 4;         // 0: lanes 0-15, 1: lanes 16-31
    const int r    = lane & 15;

    // pair info for this wave's 16-row tile (overlaps the async staging)
    const int idxInSeg = (lb * 8 + wave) * TILE + r;
    int tok = 0; float gte = 0.0f;
    if (idxInSeg < segCnt) { tok = pairTok[segBase + idxInSeg]; gte = pairGate[segBase + idxInSeg]; }
    if (lane < 16) { stok[wave][r] = tok; sg[wave][r] = gte; }

    stage_wait();
    __syncthreads();

    // ---- fc1: H(16x64) = X(16x256) @ W1e(256x64) + b1 ----
    v8f acc[4];
#pragma unroll
    for (int n = 0; n < 4; ++n) {
        const float bv = b1e[n * 16 + r];
#pragma unroll
        for (int v = 0; v < 8; ++v) acc[n][v] = bv;
    }
    // A layout: lane holds row r, K = {4k+2*half, 4k+2*half+1}
    const float* xrow = x + (size_t)tok * EMBED + 2 * half;
#pragma unroll 4
    for (int k = 0; k < 64; ++k) {
        const v2f a = *(const v2f*)(xrow + 4 * k);
        const int d = 4 * k + 2 * half;
#pragma unroll
        for (int n = 0; n < 4; ++n) {
            const v2f b = *(const v2f*)&sW1T[(n * 16 + r) * W1T_STRIDE + d];
            acc[n] = __builtin_amdgcn_wmma_f32_16x16x4_f32(
                false, a, false, b, (short)0, acc[n], false, false);
        }
    }

    // relu, spill H to LDS (C/D layout: row = v + 8*half, col = n*16+r)
    float* hb = hbuf[wave];
#pragma unroll
    for (int n = 0; n < 4; ++n)
#pragma unroll
        for (int v = 0; v < 8; ++v) {
            const float hv = acc[n][v] > 0.0f ? acc[n][v] : 0.0f;
            hb[(v + 8 * half) * HB_STRIDE + n * 16 + r] = hv;
        }
    __syncthreads();

    float gv[8]; int tv[8];
#pragma unroll
    for (int v = 0; v < 8; ++v) { gv[v] = sg[wave][v + 8 * half]; tv[v] = stok[wave][v + 8 * half]; }

    // ---- fc2: Y(16x256) = H(16x64) @ W2e(64x256) + b2, gate-scale, atomic combine ----
    for (int n2 = 0; n2 < 16; ++n2) {
        v8f c2;
        const float bv2 = b2e[n2 * 16 + r];
#pragma unroll
        for (int v = 0; v < 8; ++v) c2[v] = bv2;
#pragma unroll
        for (int kk = 0; kk < 16; ++kk) {
            const v2f a2 = *(const v2f*)&hb[r * HB_STRIDE + 4 * kk + 2 * half];
            const v2f bb = *(const v2f*)&sW2T[(n2 * 16 + r) * W2T_STRIDE + 4 * kk + 2 * half];
            c2 = __builtin_amdgcn_wmma_f32_16x16x4_f32(
                false, a2, false, bb, (short)0, c2, false, false);
        }
#pragma unroll
        for (int v = 0; v < 8; ++v) {
            const float val = c2[v] * gv[v];           // padded rows: g == 0 -> adds 0
            atomicAdd(out + (size_t)tv[v] * EMBED + n2 * 16 + r, val);
        }
    }
}

// ---------------- |y| partial reduce + in-place 0.5 scale ----------------
__global__ __launch_bounds__(256) void moe_kd1(float* __restrict__ out,
                                               float* __restrict__ partials) {
    __shared__ float red[256];
    const size_t base = (size_t)blockIdx.x * 4096;
    float s = 0.0f;
#pragma unroll
    for (int i = 0; i < 16; ++i) {
        const size_t idx = base + threadIdx.x + (size_t)i * 256;
        const float v = out[idx];
        s += fabsf(v);
        out[idx] = v * GATE_SCALE;
    }
    red[threadIdx.x] = s;
    __syncthreads();
    for (int off = 128; off > 0; off >>= 1) {
        if (threadIdx.x < off) red[threadIdx.x] += red[threadIdx.x + off];
        __syncthreads();
    }
    if (threadIdx.x == 0) partials[blockIdx.x] = red[0];
}

__global__ __launch_bounds__(256) void moe_kd2(const float* __restrict__ partials,
                                               const float* __restrict__ auxLoss,
                                               float* __restrict__ loss) {
    __shared__ float red[256];
    float s = 0.0f;
#pragma unroll
    for (int i = 0; i < 16; ++i) s += partials[threadIdx.x + i * 256];
    red[threadIdx.x] = s;
    __syncthreads();
    for (int off = 128; off > 0; off >>= 1) {
        if (threadIdx.x < off) red[threadIdx.x] += red[threadIdx.x + off];
        __syncthreads();
    }
    if (threadIdx.x == 0)
        *loss = *auxLoss + red[0] / (float)((size_t)T_TOT * EMBED);
}

// ---------------- host launcher ----------------
extern "C" void kernel_launch(void* const* d_in, const int* in_sizes, int n_in,
                              void* d_out, int out_size, void* d_ws, size_t ws_size,
                              hipStream_t stream) {
    const float* x     = (const float*)d_in[0];
    const float* wgate = (const float*)d_in[1];
    const float* w1    = (const float*)d_in[2];
    const float* b1    = (const float*)d_in[3];
    const float* w2    = (const float*)d_in[4];
    const float* b2    = (const float*)d_in[5];
    float* out = (float*)d_out;

    float* wsf = (float*)d_ws;
    int*   wsi = (int*)d_ws;
    float* importance = wsf + 0;
    int*   loadCnt    = wsi + 16;
    int*   offsets    = wsi + 32;
    int*   blkOff     = wsi + 48;
    int*   cursors    = wsi + 64;
    float* auxLoss    = wsf + 80;
    int*   nBlocks    = wsi + 81;
    int*   tmpIdx     = wsi + 128;
    float* tmpGate    = wsf + 128 + NPAIR;
    int*   pairTok    = wsi + 128 + 2 * NPAIR;
    float* pairGate   = wsf + 128 + 3 * NPAIR;
    float* partials   = wsf + 128 + 4 * NPAIR;
    float* loss       = out + (size_t)T_TOT * EMBED;

    moe_zero<<<(T_TOT * EMBED) / 256, 256, 0, stream>>>(out, importance, loadCnt);
    moe_gate<<<T_TOT / 8, 256, 0, stream>>>(x, wgate, tmpIdx, tmpGate, importance, loadCnt);
    moe_scan<<<1, 1, 0, stream>>>(importance, loadCnt, offsets, blkOff, cursors, auxLoss, nBlocks);
    moe_scatter<<<NPAIR / 256, 256, 0, stream>>>(tmpIdx, tmpGate, offsets, cursors, pairTok, pairGate);
    moe_ffn<<<BLOCKS_UB, 256, 0, stream>>>(x, w1, b1, w2, b2, pairTok, pairGate,
                                           offsets, blkOff, loadCnt, nBlocks, out);
    moe_kd1<<<(T_TOT * EMBED) / 4096, 256, 0, stream>>>(out, partials);
    moe_kd2<<<1, 256, 0, stream>>>(partials, auxLoss, loss);
}